// ROIPooling_9869834846839
// MI455X (gfx1250) — compile-verified
//
#include <hip/hip_runtime.h>
#include <stdint.h>

#define POOL 7
#define CCH  512
#define HH   50
#define WW   50
#define NC   16    // channels staged per block (LDS tile = NC*50*50*4B = 160,000 B; 2 blocks/WGP)

typedef unsigned int v4u __attribute__((ext_vector_type(4)));
typedef int          v8i __attribute__((ext_vector_type(8)));
typedef int          v4i __attribute__((ext_vector_type(4)));

__global__ __launch_bounds__(256) void roi_pool_tdm_kernel(
    const float* __restrict__ feat,   // (512, 50, 50)
    const int*   __restrict__ roi,    // (N, 4) int32
    float*       __restrict__ out)    // (N, 512, 7, 7)
{
    __shared__ float smem[NC * HH * WW];

    const int box = blockIdx.x;
    const int c0  = blockIdx.y * NC;

    // floor(coord * 1/16) for non-negative ints == >> 4
    const int x1 = roi[box * 4 + 0] >> 4;
    const int y1 = roi[box * 4 + 1] >> 4;
    const int x2 = roi[box * 4 + 2] >> 4;
    const int y2 = roi[box * 4 + 3] >> 4;

    int lenX = min(x2 + 1, WW) - x1; lenX = max(lenX, 1);
    int lenY = min(y2 + 1, HH) - y1; lenY = max(lenY, 1);

    // One wave per block issues the TDM load of the (NC, lenY, lenX) tile.
    // Scalar (SGPR) condition -> s_cbranch, so other waves never issue the
    // tensor op (TDM ignores EXEC, so EXEC-masking would not suppress it).
    const int wave = __builtin_amdgcn_readfirstlane((int)(threadIdx.x >> 5));
    if (wave == 0) {
        const uint64_t gaddr = (uint64_t)(uintptr_t)feat +
            4ull * ((uint64_t)c0 * (HH * WW) + (uint64_t)y1 * WW + (uint64_t)x1);
        const uint32_t lds_addr = (uint32_t)(size_t)(&smem[0]); // low 32 bits of LDS aperture addr = LDS offset

        // ---- D# group 0 (128b): count=1 | lds_addr | global_addr[56:0] | type=2
        v4u g0;
        g0[0] = 1u;                                   // count=1 (valid), user mode
        g0[1] = lds_addr;                             // lds_addr (bytes)
        g0[2] = (uint32_t)(gaddr & 0xffffffffull);    // global_addr[31:0]
        g0[3] = (uint32_t)((gaddr >> 32) & 0x1ffffffull) | (2u << 30); // ga[56:32] | type=2

        // ---- D# group 1 (256b)
        v8i g1;
        g1[0] = (2 << 16);                            // wg_mask=0 | data_size=2 (4B) | no flags
        g1[1] = (WW & 0xffff) << 16;                  // atomic_barrier=0 | tensor_dim0[15:0]
        g1[2] = ((WW >> 16) & 0xffff) | ((HH & 0xffff) << 16);   // dim0[31:16] | dim1[15:0]
        g1[3] = ((HH >> 16) & 0xffff) | ((lenX & 0xffff) << 16); // dim1[31:16] | tile_dim0
        g1[4] = (lenY & 0xffff) | (NC << 16);         // tile_dim1 | tile_dim2
        g1[5] = WW;                                   // tensor_dim0_stride[31:0]  (row stride, elems)
        g1[6] = ((HH * WW) & 0xffff) << 16;           // stride0[47:32]=0 | stride1[15:0] (chan stride)
        g1[7] = (HH * WW) >> 16;                      // tensor_dim1_stride[47:16]

        // ---- D# group 2 (128b): tensor_dim2=512, no dim3/iterate
        v4i g2;
        g2[0] = CCH;   // tensor_dim2
        g2[1] = 0;     // tensor_dim3 / lds_addr_increment
        g2[2] = 0;     // tensor_dim2_stride lo / global_addr_increment
        g2[3] = 0;     // stride hi | tile_dim3=0

        // ---- D# group 3 (128b): unused for 3D
        v4i g3 = {0, 0, 0, 0};

        // ---- extra 256b operand present in the 6-arg builtin form (zeroed)
        v8i g4 = {0, 0, 0, 0, 0, 0, 0, 0};

        __builtin_amdgcn_tensor_load_to_lds(g0, g1, g2, g3, g4, 0);
        __builtin_amdgcn_s_wait_tensorcnt(0);
    }
    __syncthreads();

    // NC*49 = 784 outputs: (c_local, ph, pw) each max-reduce one bin from the
    // packed LDS tile. 256 threads -> ~3 bins per thread, output stores are a
    // single contiguous 784-float run per block.
    const int chanTile = lenY * lenX;
    for (int t = threadIdx.x; t < NC * POOL * POOL; t += 256) {
        const int cl = t / (POOL * POOL);
        const int r  = t % (POOL * POOL);
        const int ph = r / POOL;
        const int pw = r % POOL;

        // bin ranges relative to tile origin (guaranteed non-empty, width <= 8)
        const int ys = (ph * lenY) / POOL;
        const int ye = ((ph + 1) * lenY + POOL - 1) / POOL;
        const int xs = (pw * lenX) / POOL;
        const int xe = ((pw + 1) * lenX + POOL - 1) / POOL;

        const float* base = &smem[cl * chanTile];
        float m = -__builtin_inff();
        for (int y = ys; y < ye; ++y) {
            const float* row = base + y * lenX;
            for (int x = xs; x < xe; ++x)
                m = fmaxf(m, row[x]);
        }
        out[(((size_t)box * CCH + (size_t)(c0 + cl)) * POOL + ph) * POOL + pw] = m;
    }
}

extern "C" void kernel_launch(void* const* d_in, const int* in_sizes, int n_in,
                              void* d_out, int out_size, void* d_ws, size_t ws_size,
                              hipStream_t stream) {
    const float* feat = (const float*)d_in[0];   // (1, 512, 50, 50) fp32
    const int*   roi  = (const int*)d_in[1];     // (N, 4) int32
    float*       out  = (float*)d_out;           // (N, 512, 7, 7) fp32

    const int nbox = in_sizes[1] / 4;            // 1024
    dim3 grid(nbox, CCH / NC, 1);                // 1024 x 32 blocks
    dim3 block(256, 1, 1);                       // 8 waves (wave32)
    roi_pool_tdm_kernel<<<grid, block, 0, stream>>>(feat, roi, out);
}